// FineSubMatching_7421703487995
// MI455X (gfx1250) — compile-verified
//
#include <hip/hip_runtime.h>
#include <hip/hip_bf16.h>

// CDNA5 (gfx1250) fused fine-matching kernel.
//
// Fusion: f0.f1^T = feat0 (P^T P) feat1^T / 64, so Gram G = P^T P is computed
// once; each window then needs only 2 f32 WMMA GEMMs (64x64x64) instead of 3.
// Memory-bound (~33 FLOP/byte after fusion) -> FP32 WMMA (16x16x4) keeps
// reference precision at no throughput cost; data movement uses the gfx1250
// async Global->LDS path (ASYNCcnt) and non-temporal streaming stores.

typedef float v2f __attribute__((ext_vector_type(2)));
typedef float v8f __attribute__((ext_vector_type(8)));

// Padded LDS stride: 68 floats -> row pitch 272B (16B aligned for async B128),
// banks = (4*row + col) % 64: conflict-free for A/B fragment and conf access.
#define LDS_S 68

// ---- gfx1250 async Global->LDS copy (CDNA5 ISA, GVS mode) ----
// lds_byte: LDS byte offset (low 32 bits of flat LDS address == LDS offset)
// goff:     unsigned byte offset from SGPR base
__device__ __forceinline__ void async_copy_b128(unsigned lds_byte, unsigned goff,
                                                const void* sbase) {
    asm volatile("global_load_async_to_lds_b128 %0, %1, %2 offset:0"
                 :: "v"(lds_byte), "v"(goff), "s"(sbase)
                 : "memory");
}
__device__ __forceinline__ void wait_async0() {
    asm volatile("s_wait_asynccnt 0x0" ::: "memory");
}

// ---------------------------------------------------------------- Gram matrix
__global__ __launch_bounds__(256) void gram_kernel(const float* __restrict__ P,
                                                   float* __restrict__ G) {
    // G[d1][d2] = sum_d P[d][d1] * P[d][d2]   (64x64)
    int idx = blockIdx.x * 256 + threadIdx.x;   // 0..4095
    int d1 = idx >> 6, d2 = idx & 63;
    float s = 0.f;
    #pragma unroll 8
    for (int d = 0; d < 64; ++d) s += P[d * 64 + d1] * P[d * 64 + d2];
    G[idx] = s;
}

// ------------------------------------------------------------- main per-window
__global__ __launch_bounds__(256) void fine_match_kernel(
    const float* __restrict__ f0g, const float* __restrict__ f1g,
    const float* __restrict__ Gg,  const float* __restrict__ w1,
    const float* __restrict__ w2,  const int* __restrict__ i_ids,
    const int* __restrict__ j_ids,
    float* __restrict__ mk0, float* __restrict__ mk1,
    float* __restrict__ mconf, float* __restrict__ confg)
{
    __shared__ __align__(16) float f0s[64 * LDS_S];
    __shared__ __align__(16) float f1s[64 * LDS_S];
    __shared__ __align__(16) float Gs [64 * LDS_S];
    __shared__ __align__(16) float Ts [64 * LDS_S];  // T, then overlaid with sim
    __shared__ float rowmax[64], rowsum[64], colmax[64], colsum[64];
    __shared__ float redv[256];
    __shared__ int   redi[256];
    __shared__ float cat[128];
    __shared__ float hbuf[128];
    __shared__ float dlt[4];

    const int n    = blockIdx.x;
    const int t    = threadIdx.x;
    const int lane = t & 31;
    const int wave = t >> 5;

    // ---- async Global->LDS staging: feat0[n], feat1[n], G (B128 per lane)
    const float* f0p = f0g + (size_t)n * 4096;
    const float* f1p = f1g + (size_t)n * 4096;
    #pragma unroll
    for (int r = 0; r < 4; ++r) {
        int idx = (t + r * 256) * 4;            // element index, multiple of 4
        int l = idx >> 6, c = idx & 63;
        unsigned goff = (unsigned)idx * 4u;     // byte offset in 16KB tile
        unsigned loff = (unsigned)((l * LDS_S + c) * 4);
        async_copy_b128((unsigned)(uintptr_t)f0s + loff, goff, f0p);
        async_copy_b128((unsigned)(uintptr_t)f1s + loff, goff, f1p);
        async_copy_b128((unsigned)(uintptr_t)Gs  + loff, goff, Gg);
    }
    wait_async0();
    __syncthreads();

    // WMMA f32 16x16x4 fragment indexing (wave32):
    //  A: lane holds row m = lane&15; K-half 0/2 selected by lane>=16; 2 VGPRs
    //  B: lane holds col n = lane&15; same K-half split; 2 VGPRs
    //  C/D: 8 VGPRs; lane col = lane&15, row base = (lane>=16)*8
    const int ml = lane & 15;
    const int kh = (lane >> 4) << 1;     // 0 or 2
    const int mhalf = (lane >> 4) * 8;

    v8f acc[2];

    // ---- GEMM1: T = feat0 @ G     (16 tiles of 16x16, 2 per wave)
    #pragma unroll
    for (int tt = 0; tt < 2; ++tt) {
        int tile = wave * 2 + tt;
        int mt = tile >> 2, nt = tile & 3;
        v8f c = {};
        #pragma unroll
        for (int k0 = 0; k0 < 64; k0 += 4) {
            int k = k0 + kh;
            v2f a, b;
            a.x = f0s[(mt * 16 + ml) * LDS_S + k];
            a.y = f0s[(mt * 16 + ml) * LDS_S + k + 1];
            b.x = Gs[k * LDS_S + nt * 16 + ml];
            b.y = Gs[(k + 1) * LDS_S + nt * 16 + ml];
            c = __builtin_amdgcn_wmma_f32_16x16x4_f32(false, a, false, b,
                                                      (short)0, c, false, false);
        }
        acc[tt] = c;
    }
    #pragma unroll
    for (int tt = 0; tt < 2; ++tt) {
        int tile = wave * 2 + tt;
        int mt = tile >> 2, nt = tile & 3;
        int mb = mt * 16 + mhalf, nc = nt * 16 + ml;
        #pragma unroll
        for (int v = 0; v < 8; ++v) Ts[(mb + v) * LDS_S + nc] = acc[tt][v];
    }
    __syncthreads();

    // ---- GEMM2: sim = T @ feat1^T * 1/(64*0.1)
    #pragma unroll
    for (int tt = 0; tt < 2; ++tt) {
        int tile = wave * 2 + tt;
        int mt = tile >> 2, nt = tile & 3;
        v8f c = {};
        #pragma unroll
        for (int k0 = 0; k0 < 64; k0 += 4) {
            int k = k0 + kh;
            v2f a, b;
            a.x = Ts[(mt * 16 + ml) * LDS_S + k];
            a.y = Ts[(mt * 16 + ml) * LDS_S + k + 1];
            b.x = f1s[(nt * 16 + ml) * LDS_S + k];      // B = feat1^T
            b.y = f1s[(nt * 16 + ml) * LDS_S + k + 1];
            c = __builtin_amdgcn_wmma_f32_16x16x4_f32(false, a, false, b,
                                                      (short)0, c, false, false);
        }
        acc[tt] = c;
    }
    __syncthreads();                     // everyone done reading T
    const float simscale = 1.0f / 6.4f;  // 1/(sqrt(C)^2 * TEMPERATURE)
    #pragma unroll
    for (int tt = 0; tt < 2; ++tt) {
        int tile = wave * 2 + tt;
        int mt = tile >> 2, nt = tile & 3;
        int mb = mt * 16 + mhalf, nc = nt * 16 + ml;
        #pragma unroll
        for (int v = 0; v < 8; ++v)
            Ts[(mb + v) * LDS_S + nc] = acc[tt][v] * simscale;  // sim overlays T
    }
    __syncthreads();

    // ---- dual-softmax statistics: threads 0-63 rows, 64-127 cols
    if (t < 64) {
        float m = -3.4e38f;
        for (int j = 0; j < 64; ++j) m = fmaxf(m, Ts[t * LDS_S + j]);
        float s = 0.f;
        for (int j = 0; j < 64; ++j) s += __expf(Ts[t * LDS_S + j] - m);
        rowmax[t] = m; rowsum[t] = s;
    } else if (t < 128) {
        int sc = t - 64;
        float m = -3.4e38f;
        for (int l = 0; l < 64; ++l) m = fmaxf(m, Ts[l * LDS_S + sc]);
        float s = 0.f;
        for (int l = 0; l < 64; ++l) s += __expf(Ts[l * LDS_S + sc] - m);
        colmax[sc] = m; colsum[sc] = s;
    }
    __syncthreads();

    // ---- conf = softmax_rows * softmax_cols (single exp), streaming NT store,
    //      fused local argmax (first-occurrence tie-break)
    float bv = -1.f; int bi = 0;
    float* confn = confg + (size_t)n * 4096;
    #pragma unroll
    for (int r = 0; r < 16; ++r) {
        int idx = t + r * 256;
        int l = idx >> 6, sc = idx & 63;
        float x = Ts[l * LDS_S + sc];
        float cval = __expf(2.f * x - rowmax[l] - colmax[sc]) /
                     (rowsum[l] * colsum[sc]);
        __builtin_nontemporal_store(cval, &confn[idx]);   // 134MB write-once
        if (cval > bv) { bv = cval; bi = idx; }
    }
    redv[t] = bv; redi[t] = bi;
    __syncthreads();
    for (int off = 128; off > 0; off >>= 1) {
        if (t < off) {
            float v2 = redv[t + off]; int i2 = redi[t + off];
            if (v2 > redv[t] || (v2 == redv[t] && i2 < redi[t])) {
                redv[t] = v2; redi[t] = i2;
            }
        }
        __syncthreads();
    }
    const int   best  = redi[0];
    const float bestv = redv[0];
    const int il = best >> 6, jl = best & 63;

    // ---- tiny MLP on concatenated raw features of the best cell
    if (t < 128) cat[t] = (t < 64) ? f0s[il * LDS_S + t]
                                   : f1s[jl * LDS_S + (t - 64)];
    __syncthreads();
    if (t < 128) {
        float s = 0.f;
        const float* wr = w1 + t * 128;
        #pragma unroll 8
        for (int j = 0; j < 128; ++j) s += cat[j] * wr[j];
        hbuf[t] = fmaxf(s, 0.f);
    }
    __syncthreads();
    if (t < 4) {
        float s = 0.f;
        const float* wr = w2 + t * 128;
        #pragma unroll 8
        for (int j = 0; j < 128; ++j) s += hbuf[j] * wr[j];
        dlt[t] = tanhf(s) * 0.5f;
    }
    __syncthreads();

    // ---- coordinates (W=8 even -> pad=0; SCALE_F2C=4, SCALE_I2F=2)
    if (t == 0) {
        int ii = i_ids[n], jj = j_ids[n];
        float a0x = (float)((ii % 160) * 4), a0y = (float)((ii / 160) * 4);
        float a1x = (float)((jj % 160) * 4), a1y = (float)((jj / 160) * 4);
        float w0x = (float)(il & 7), w0y = (float)(il >> 3);
        float w1x = (float)(jl & 7), w1y = (float)(jl >> 3);
        mk0[2 * n]     = (w0x + a0x + dlt[0]) * 2.0f;
        mk0[2 * n + 1] = (w0y + a0y + dlt[1]) * 2.0f;
        mk1[2 * n]     = (w1x + a1x + dlt[2]) * 2.0f;
        mk1[2 * n + 1] = (w1y + a1y + dlt[3]) * 2.0f;
        mconf[n] = (bestv > 0.1f) ? bestv : 0.0f;
    }
}

// ---------------------------------------------------------------------- launch
extern "C" void kernel_launch(void* const* d_in, const int* in_sizes, int n_in,
                              void* d_out, int out_size, void* d_ws, size_t ws_size,
                              hipStream_t stream) {
    const float* f0   = (const float*)d_in[0];
    const float* f1   = (const float*)d_in[1];
    const float* proj = (const float*)d_in[2];
    const float* w1   = (const float*)d_in[3];
    const float* w2   = (const float*)d_in[4];
    // d_in[5] = b_ids (unused by the reference computation)
    const int* i_ids  = (const int*)d_in[6];
    const int* j_ids  = (const int*)d_in[7];

    const int M = in_sizes[5];          // 8192 windows
    float* out  = (float*)d_out;
    float* mk0  = out;                  // (M,2)
    float* mk1  = out + 2 * (size_t)M;  // (M,2)
    float* mc   = out + 4 * (size_t)M;  // (M,)
    float* conf = out + 5 * (size_t)M;  // (M,64,64)

    float* G = (float*)d_ws;            // 64x64 Gram matrix scratch (16 KB)

    gram_kernel<<<16, 256, 0, stream>>>(proj, G);
    fine_match_kernel<<<M, 256, 0, stream>>>(f0, f1, G, w1, w2, i_ids, j_ids,
                                             mk0, mk1, mc, conf);
}